// TrajectoryConsistencyLoss_19774029431354
// MI455X (gfx1250) — compile-verified
//
#include <hip/hip_runtime.h>

#define T_LEN   256
#define B_N     2048
#define D_DIM   64
#define XSTRIDE 66          // padded row stride (floats) -> bank-conflict-free both phases
#define MAX_ITER_K 50
#define STEP_K  0.1f
#define EPSF    1e-8f
#define MAXN    (1.0f - 1e-8f)
#define RADIUS_K 2.0f

typedef __attribute__((ext_vector_type(2))) float v2f;
typedef __attribute__((ext_vector_type(8))) float v8f;

__device__ __forceinline__ float waveSum(float v) {
#pragma unroll
  for (int m = 16; m >= 1; m >>= 1) v += __shfl_xor(v, m, 32);
  return v;
}

__device__ __forceinline__ v8f wmma4(v2f a, v2f b, v8f c) {
  // D(16x16) = A(16x4) * B(4x16) + C, fp32 throughout
  return __builtin_amdgcn_wmma_f32_16x16x4_f32(false, a, false, b, (short)0, c,
                                               false, false);
}

// q[t] = X_t . c for all 256 rows. Wave `wid` handles row blocks 2*wid, 2*wid+1.
// B columns all replicate c -> every column of D equals the q block.
__device__ __forceinline__ void matvec_q(const float* __restrict__ Xl,
                                         const float* __restrict__ cs,
                                         float* __restrict__ qs,
                                         int wid, int lane) {
  const int m = lane & 15;
  const int hi = lane >> 4;
  const int kofs = hi * 2;
#pragma unroll
  for (int rbi = 0; rbi < 2; ++rbi) {
    const int rb = wid * 2 + rbi;
    const float* xrow = Xl + (rb * 16 + m) * XSTRIDE;
    v8f acc = {0.f, 0.f, 0.f, 0.f, 0.f, 0.f, 0.f, 0.f};
#pragma unroll
    for (int k0 = 0; k0 < D_DIM; k0 += 4) {
      v2f a, b;
      a.x = xrow[k0 + kofs];
      a.y = xrow[k0 + kofs + 1];
      b.x = cs[k0 + kofs];          // independent of (lane&15): replicate over N
      b.y = cs[k0 + kofs + 1];
      acc = wmma4(a, b, acc);
    }
    if (m == 0) {
      const int base = rb * 16 + hi * 8;   // C/D layout: VGPR j -> M=j (+8 for hi lanes)
#pragma unroll
      for (int j = 0; j < 8; ++j) qs[base + j] = acc[j];
    }
  }
}

// Sp[wid][0..15] = sum over a 128-row half of w_t * X_t[d0..d0+15].
// A rows all replicate w -> every row of D equals the partial sum.
__device__ __forceinline__ void rowsum_w(const float* __restrict__ Xl,
                                         const float* __restrict__ wsh,
                                         float* __restrict__ Sp,
                                         int wid, int lane) {
  const int m = lane & 15;
  const int hi = lane >> 4;
  const int kofs = hi * 2;
  const int d0 = (wid & 3) * 16;
  const int tb = (wid >> 2) * 128;
  v8f acc = {0.f, 0.f, 0.f, 0.f, 0.f, 0.f, 0.f, 0.f};
  for (int t0 = tb; t0 < tb + 128; t0 += 4) {
    v2f a, b;
    a.x = wsh[t0 + kofs];           // independent of (lane&15): replicate over M
    a.y = wsh[t0 + kofs + 1];
    b.x = Xl[(t0 + kofs) * XSTRIDE + d0 + m];
    b.y = Xl[(t0 + kofs + 1) * XSTRIDE + d0 + m];
    acc = wmma4(a, b, acc);
  }
  if (hi == 0) Sp[wid * 16 + m] = acc[0];   // row M=0, N=lane
}

__global__ __launch_bounds__(256) void frechet_loss_kernel(
    const float* __restrict__ Xg, float* __restrict__ losses) {
  extern __shared__ float smem[];
  float* Xl    = smem;                       // 256*66
  float* y2s   = Xl + T_LEN * XSTRIDE;       // 256
  float* wsh   = y2s + T_LEN;                // 256
  float* qs    = wsh + T_LEN;                // 256
  float* cs    = qs + T_LEN;                 // 64
  float* grads = cs + D_DIM;                 // 64
  float* Sp    = grads + D_DIM;              // 8*16
  float* pr8   = Sp + 128;                   // 8
  float* scal  = pr8 + 8;                    // 8: [0]=c2 [1]=G [3]=done

  const int tid  = threadIdx.x;
  const int lane = tid & 31;
  const int wid  = tid >> 5;
  const int b    = blockIdx.x;

  // ---- stage + project rows; y2 and init log0 weights ----
  for (int rr = 0; rr < 32; ++rr) {
    const int r = wid * 32 + rr;
    const float2 xv = reinterpret_cast<const float2*>(
        Xg + ((size_t)r * B_N + b) * D_DIM)[lane];
    float x0 = xv.x, x1 = xv.y;
    const float n2 = waveSum(x0 * x0 + x1 * x1);
    const float n  = sqrtf(n2);
    const float pr = fminf(MAXN / (n + EPSF), 1.0f);
    x0 *= pr; x1 *= pr;
    Xl[r * XSTRIDE + 2 * lane]     = x0;
    Xl[r * XSTRIDE + 2 * lane + 1] = x1;
    if (lane == 0) {
      const float y2 = pr * pr * n2;
      y2s[r] = y2;
      const float ny = pr * n;
      const float z  = fminf(ny, 1.0f - 1e-4f);
      const float lc = atanhf(z) / (ny + EPSF);
      wsh[r] = (ny < EPSF) ? 0.0f : lc;       // log0 row weight
    }
  }
  __syncthreads();

  // ---- center init: project(exp0(mean_t log0(X_t))) ----
  rowsum_w(Xl, wsh, Sp, wid, lane);
  __syncthreads();
  if (tid < D_DIM)
    grads[tid] = (Sp[(tid >> 4) * 16 + (tid & 15)] +
                  Sp[((tid >> 4) + 4) * 16 + (tid & 15)]) * (1.0f / T_LEN);
  __syncthreads();
  if (wid == 0) {
    const float m1 = grads[lane], m2 = grads[lane + 32];
    const float n  = sqrtf(waveSum(m1 * m1 + m2 * m2));
    const float co = tanhf(n) / (n + EPSF);
    const float ps = fminf(MAXN / (co * n + EPSF), 1.0f);
    const float sc = (n < EPSF) ? 0.0f : co * ps;
    const float c1 = sc * m1, c2e = sc * m2;
    cs[lane] = c1; cs[lane + 32] = c2e;
    const float c2 = waveSum(c1 * c1 + c2e * c2e);
    if (lane == 0) scal[0] = c2;
  }
  __syncthreads();

  // ---- Riemannian GD iterations ----
  for (int it = 0; it < MAX_ITER_K; ++it) {
    matvec_q(Xl, cs, qs, wid, lane);
    __syncthreads();
    {
      const float q  = qs[tid];
      const float y2 = y2s[tid];
      const float c2 = scal[0];
      const float a   = 1.0f - 2.0f * q + y2;       // coeff on x = -c
      const float bb  = 1.0f - c2;                  // coeff on y = X_t
      const float den = fmaxf(1.0f - 2.0f * q + c2 * y2, EPSF);
      const float u2  = fmaxf(a * a * c2 - 2.0f * a * bb * q + bb * bb * y2, 0.0f)
                        / (den * den);
      const float nu  = sqrtf(u2);
      const float ps  = fminf(MAXN / (nu + EPSF), 1.0f);   // project(num/den)
      const float n2  = ps * nu;
      const float z   = fminf(n2, 1.0f - 1e-4f);
      const float lc  = atanhf(z) / (n2 + EPSF);
      const float f   = (n2 < EPSF) ? 0.0f : lc * ps / den;
      wsh[tid] = f * bb;                            // weight on X_t
      const float gw = waveSum(-f * a);             // weight on c (partial)
      if (lane == 0) pr8[wid] = gw;
    }
    __syncthreads();
    rowsum_w(Xl, wsh, Sp, wid, lane);
    if (tid == 0) {
      float s = 0.f;
#pragma unroll
      for (int i = 0; i < 8; ++i) s += pr8[i];
      scal[1] = s;
    }
    __syncthreads();
    if (tid < D_DIM) {
      const float S = Sp[(tid >> 4) * 16 + (tid & 15)] +
                      Sp[((tid >> 4) + 4) * 16 + (tid & 15)];
      grads[tid] = (scal[1] * cs[tid] + S) * (1.0f / T_LEN);
    }
    __syncthreads();
    if (wid == 0) {
      const float g1 = grads[lane], g2 = grads[lane + 32];
      const float gn = sqrtf(waveSum(g1 * g1 + g2 * g2));
      const bool done = gn < 1e-6f;                 // latch BEFORE update
      if (!done) {
        const float c2 = scal[0];
        const float n  = STEP_K * gn;               // ||-STEP*grad||
        const float co = tanhf(n) / (n + EPSF);
        const float ps = fminf(MAXN / (co * n + EPSF), 1.0f);
        const float sc = (n < EPSF) ? 0.0f : co * ps;
        const float e1 = sc * (-STEP_K * g1), e2 = sc * (-STEP_K * g2);
        const float e2n = waveSum(e1 * e1 + e2 * e2);
        const float c1 = cs[lane], cc2 = cs[lane + 32];
        const float xy = waveSum(c1 * e1 + cc2 * e2);
        const float aa  = 1.0f + 2.0f * xy + e2n;
        const float bb2 = 1.0f - c2;
        const float den = fmaxf(1.0f + 2.0f * xy + c2 * e2n, EPSF);
        const float u1 = (aa * c1 + bb2 * e1) / den;
        const float u2 = (aa * cc2 + bb2 * e2) / den;
        const float un = sqrtf(waveSum(u1 * u1 + u2 * u2));
        const float pu = fminf(MAXN / (un + EPSF), 1.0f);
        const float nc1 = pu * u1, nc2 = pu * u2;
        cs[lane] = nc1; cs[lane + 32] = nc2;
        const float nc2s = waveSum(nc1 * nc1 + nc2 * nc2);
        if (lane == 0) scal[0] = nc2s;
      }
      if (lane == 0) scal[3] = done ? 1.0f : 0.0f;
    }
    __syncthreads();
    if (scal[3] != 0.0f) break;                     // workgroup-uniform
  }

  // ---- distances + hinge mean ----
  __syncthreads();
  matvec_q(Xl, cs, qs, wid, lane);
  __syncthreads();
  {
    const float q  = qs[tid];
    const float y2 = y2s[tid];
    const float c2 = scal[0];
    const float diff2 = fmaxf(y2 - 2.0f * q + c2, 0.0f);
    const float denom = fmaxf((1.0f - y2) * (1.0f - c2), EPSF);
    const float z     = fmaxf(1.0f + 2.0f * diff2 / denom, 1.0f + 1e-6f);
    const float viol  = fmaxf(acoshf(z) - RADIUS_K, 0.0f);
    const float s = waveSum(viol);
    if (lane == 0) pr8[wid] = s;
  }
  __syncthreads();
  if (tid == 0) {
    float s = 0.f;
#pragma unroll
    for (int i = 0; i < 8; ++i) s += pr8[i];
    losses[b] = s * (1.0f / T_LEN);
  }
}

__global__ __launch_bounds__(256) void reduce_mean_kernel(
    const float* __restrict__ losses, float* __restrict__ out, int n) {
  __shared__ float sh[256];
  float s = 0.f;
  for (int i = threadIdx.x; i < n; i += 256) s += losses[i];
  sh[threadIdx.x] = s;
  __syncthreads();
  for (int st = 128; st > 0; st >>= 1) {
    if (threadIdx.x < st) sh[threadIdx.x] += sh[threadIdx.x + st];
    __syncthreads();
  }
  if (threadIdx.x == 0) out[0] = sh[0] / (float)n;
}

extern "C" void kernel_launch(void* const* d_in, const int* in_sizes, int n_in,
                              void* d_out, int out_size, void* d_ws, size_t ws_size,
                              hipStream_t stream) {
  const float* X = (const float*)d_in[0];   // [T=256, B=2048, D=64] fp32
  float* out = (float*)d_out;               // scalar fp32
  float* losses = (float*)d_ws;             // 2048 fp32 scratch
  const size_t smem_bytes =
      (size_t)(T_LEN * XSTRIDE + 3 * T_LEN + 2 * D_DIM + 128 + 16) * sizeof(float);
  frechet_loss_kernel<<<B_N, 256, smem_bytes, stream>>>(X, losses);
  reduce_mean_kernel<<<1, 256, 0, stream>>>(losses, out, B_N);
}